// A5ExactScanPlugin_64922725646541
// MI455X (gfx1250) — compile-verified
//
#include <hip/hip_runtime.h>
#include <hip/hip_bf16.h>
#include <stdint.h>

#define B_DIM 8192
#define T_DIM 2048
#define N_DIM 60
#define ROWS_PER_BLOCK 8
#define THREADS 256
#define CHUNKS_PER_ROW 32          // one wave (32 lanes) == one row's 32 chunks
#define CHUNK_LEN 64               // T / CHUNKS_PER_ROW
#define CHUNK_STRIDE 68            // 64 data dwords + 4 pad dwords (16B aligned, bank skew)

// ---- CDNA5 async-tensor path helpers -------------------------------------
#if defined(__has_builtin)
#  if __has_builtin(__builtin_amdgcn_s_wait_asynccnt)
#    define WAIT_ASYNC0() __builtin_amdgcn_s_wait_asynccnt(0)
#  endif
#endif
#ifndef WAIT_ASYNC0
#  define WAIT_ASYNC0() asm volatile("s_wait_asynccnt 0" ::: "memory")
#endif

// Low 32 bits of a generic LDS pointer == LDS byte offset (aperture base has
// zero low bits; HW adds the wave's LDS_BASE for async-to-LDS like DS ops).
__device__ __forceinline__ unsigned lds_addr32(const void* p) {
    return (unsigned)(unsigned long long)p;
}

// GVS-mode async copy: copies 16B from global (SGPR base plus 32-bit VGPR offset) to LDS at lds_off.
__device__ __forceinline__ void async_b128(unsigned lds_off, unsigned goff,
                                           const void* sbase) {
    asm volatile("global_load_async_to_lds_b128 %0, %1, %2"
                 :: "v"(lds_off), "v"(goff), "s"(sbase)
                 : "memory");
}
// --------------------------------------------------------------------------

__global__ __launch_bounds__(THREADS) void
A5ExactScanPlugin_64922725646541_kernel(const float* __restrict__ scale,
                                        const int* __restrict__ ids,
                                        const int* __restrict__ mul,
                                        float* __restrict__ out) {
    __shared__ __align__(16) int tile[THREADS * CHUNK_STRIDE];   // 69,632 B
    __shared__ __align__(16) int mulLds[N_DIM * N_DIM];          // 14,400 B
    __shared__ int finals[ROWS_PER_BLOCK];

    const int tid  = threadIdx.x;
    const int lane = tid & 31;          // chunk-in-row
    const int wave = tid >> 5;          // row-in-block
    const int row0 = blockIdx.x * ROWS_PER_BLOCK;

    // ---- Stage 1: async global -> LDS copies (ASYNCcnt path) -------------
    // Block's 8 rows are contiguous in global memory: 8*2048*4B = 64KB.
    const int* slab = ids + (size_t)row0 * T_DIM;
    const unsigned tileBase = lds_addr32(tile);
    // 4096 x 16B pieces; piece p -> chunk p>>4, sub-piece p&15.
    // Global offset p*16 is contiguous & coalesced; LDS is chunk-skewed.
    #pragma unroll
    for (int it = 0; it < 16; ++it) {
        int p = tid + it * THREADS;
        int chunk = p >> 4, piece = p & 15;
        async_b128(tileBase + (unsigned)(chunk * (CHUNK_STRIDE * 4) + piece * 16),
                   (unsigned)(p * 16), slab);
    }
    // Cayley table: 14400 B = 900 x 16B pieces.
    const unsigned mulBase = lds_addr32(mulLds);
    for (int p = tid; p < 900; p += THREADS) {
        async_b128(mulBase + (unsigned)(p * 16), (unsigned)(p * 16), mul);
    }
    WAIT_ASYNC0();
    __syncthreads();

    // ---- Stage 2: per-chunk group product (64 dependent LDS gathers) -----
    // thread tid == chunkId; chunk covers T-columns [lane*64, lane*64+64).
    int s = 0;  // identity element id
    const int* cp = &tile[tid * CHUNK_STRIDE];
    #pragma unroll
    for (int i = 0; i < CHUNK_LEN; i += 4) {
        int4 x4 = *reinterpret_cast<const int4*>(cp + i);
        s = mulLds[x4.x * N_DIM + s];
        s = mulLds[x4.y * N_DIM + s];
        s = mulLds[x4.z * N_DIM + s];
        s = mulLds[x4.w * N_DIM + s];
    }

    // ---- Stage 3: ordered wave-level tree combine (non-commutative safe) --
    // Lane l holds Q_l; want Q_31 * ... * Q_0. 'up' is the LATER segment, so
    // acc = mul[up, acc] preserves T order. 5 gathers instead of 31.
    int acc = s;
    #pragma unroll
    for (int off = 1; off < CHUNKS_PER_ROW; off <<= 1) {
        int up = __shfl_down(acc, off, 32);
        if (lane + off < CHUNKS_PER_ROW)
            acc = mulLds[up * N_DIM + acc];
    }
    if (lane == 0) finals[wave] = acc;
    __syncthreads();

    // ---- Stage 4: write +-10 one-hot logits ------------------------------
    const float sc = scale[0];
    for (int j = tid; j < ROWS_PER_BLOCK * N_DIM; j += THREADS) {
        int rr = j / N_DIM, nn = j - rr * N_DIM;
        out[(size_t)(row0 + rr) * N_DIM + nn] =
            sc * ((nn == finals[rr]) ? 10.0f : -10.0f);
    }
}

extern "C" void kernel_launch(void* const* d_in, const int* in_sizes, int n_in,
                              void* d_out, int out_size, void* d_ws, size_t ws_size,
                              hipStream_t stream) {
    (void)in_sizes; (void)n_in; (void)out_size; (void)d_ws; (void)ws_size;
    const float* scale = (const float*)d_in[0];
    const int*   ids   = (const int*)d_in[1];
    const int*   mul   = (const int*)d_in[2];
    float*       out   = (float*)d_out;

    dim3 grid(B_DIM / ROWS_PER_BLOCK);   // 1024 blocks
    dim3 block(THREADS);                 // 8 waves (wave32)
    A5ExactScanPlugin_64922725646541_kernel<<<grid, block, 0, stream>>>(scale, ids, mul, out);
}